// symmLayer_48653389529229
// MI455X (gfx1250) — compile-verified
//
#include <hip/hip_runtime.h>

// ---------------- problem constants (match reference) ----------------
#define B_ROWS   4096
#define N_PTS    8192
#define K_TOT    (N_PTS * 3)        // 24576 : reduction length
#define KS       32                 // K-split factor
#define K_CHUNK  (K_TOT / KS)       // 768 (multiple of 32)
#define M_TILES  (B_ROWS / 16)      // 256
#define WAVES_PER_BLOCK 8           // 256 threads, wave32

typedef __attribute__((ext_vector_type(2))) float v2f;
typedef __attribute__((ext_vector_type(8))) float v8f;

// ---------------------------------------------------------------------
// Kernel 1: build the 24576x3 matrix A (zero-padded to 16 cols) from
// weight[8192,6].  A[3n+i, j] = (M^T M)[j][i], M upper-tri w/ softplus diag.
// Stored K-pair interleaved: bpack[(k>>1)*32 + col*2 + (k&1)]
// so a GEMM lane's two B values (K=k, k+1 for its column) are one float2.
// ---------------------------------------------------------------------
__global__ void __launch_bounds__(256)
prep_kernel(const float* __restrict__ w, float* __restrict__ bpack) {
    int n = blockIdx.x * blockDim.x + threadIdx.x;
    if (n >= N_PTS) return;

    float w0 = w[n * 6 + 0], w1 = w[n * 6 + 1], w2 = w[n * 6 + 2];
    float w3 = w[n * 6 + 3], w4 = w[n * 6 + 4], w5 = w[n * 6 + 5];

    // numerically-stable softplus: max(x,0) + log1p(exp(-|x|))
    float a = fmaxf(w0, 0.0f) + log1pf(__expf(-fabsf(w0)));
    float b = fmaxf(w3, 0.0f) + log1pf(__expf(-fabsf(w3)));
    float c = fmaxf(w5, 0.0f) + log1pf(__expf(-fabsf(w5)));

    // wBlock = M^T M  (symmetric 3x3)
    float W[3][3];
    W[0][0] = a * a;      W[0][1] = a * w1;              W[0][2] = a * w2;
    W[1][0] = W[0][1];    W[1][1] = w1 * w1 + b * b;     W[1][2] = w1 * w2 + b * w4;
    W[2][0] = W[0][2];    W[2][1] = W[1][2];             W[2][2] = w2 * w2 + w4 * w4 + c * c;

    #pragma unroll
    for (int i = 0; i < 3; ++i) {
        int k = 3 * n + i;
        int base = (k >> 1) * 32 + (k & 1);
        #pragma unroll
        for (int col = 0; col < 16; ++col) {
            bpack[base + col * 2] = (col < 3) ? W[col][i] : 0.0f;
        }
    }
}

// ---------------------------------------------------------------------
// Kernel 2: streaming GEMM via V_WMMA_F32_16X16X4_F32.
// One wave per (m_tile, k_chunk).  Per 4-K step:
//   A frag (16x4 f32): lane L -> row = m_tile*16 + (L&15),
//                      K = k + 2*(L>>4) + {0,1}   (one float2 load)
//   B frag (4x16 f32): lane L -> col = L&15,
//                      K = k + 2*(L>>4) + {0,1}   (one float2 load, coalesced)
// Both streams advance linearly -> explicit bumped pointers so loads become
// immediate-offset clauses.  Wave-uniform indices are readfirstlane'd into
// SGPRs so loop bookkeeping / prefetch guard are scalar.
// ---------------------------------------------------------------------
__global__ void __launch_bounds__(256)
gemm_kernel(const float* __restrict__ x,
            const float* __restrict__ bpack,
            float* __restrict__ partial) {
    const int lane = threadIdx.x & 31;
    // wave-uniform scalars
    const int wave = __builtin_amdgcn_readfirstlane(
        blockIdx.x * WAVES_PER_BLOCK + (threadIdx.x >> 5));
    const int mt = wave % M_TILES;     // M tile index   (SGPR)
    const int kc = wave / M_TILES;     // K chunk index  (SGPR)
    const int k0 = kc * K_CHUNK;       // (SGPR)

    const int half = lane >> 4;        // 0: K pair {0,1}, 1: K pair {2,3}
    const int col  = lane & 15;
    const int row  = mt * 16 + col;

    const float* aptr = x + (size_t)row * K_TOT + 2 * half + k0;
    const float* bptr = bpack + col * 2 + ((size_t)(k0 >> 1) + half) * 32;

    v8f acc = {};

    for (int kk = 0; kk < K_CHUNK; kk += 32) {
        // prefetch the A stream ~1KB (256 floats) ahead; scalar guard
        if (k0 + kk + 288 <= K_TOT)
            __builtin_prefetch(aptr + 256, 0, 1);

        #pragma unroll
        for (int u = 0; u < 8; ++u) {
            v2f afrag = *(const v2f*)(aptr + 4 * u);    //  16 B stride
            v2f bfrag = *(const v2f*)(bptr + 64 * u);   // 256 B stride
            // D = A(16x4) * B(4x16) + C   -- fp32 matrix pipe
            acc = __builtin_amdgcn_wmma_f32_16x16x4_f32(
                /*neg_a=*/false, afrag,
                /*neg_b=*/false, bfrag,
                /*c_mod=*/(short)0, acc,
                /*reuse_a=*/false, /*reuse_b=*/false);
        }
        aptr += 32;        //  32 floats = 8 K-steps of A
        bptr += 8 * 64;    // 512 floats = 8 K-steps of B
    }

    // C/D layout: lane L holds N = L&15; VGPR v holds M = v + 8*(L>>4).
    if (col < 3) {
        float* p = partial + ((size_t)kc * B_ROWS + mt * 16 + 8 * half) * 3 + col;
        #pragma unroll
        for (int v = 0; v < 8; ++v)
            p[v * 3] = acc[v];
    }
}

// ---------------------------------------------------------------------
// Kernel 3: sum the KS partials, add N_PTS * bias.
// ---------------------------------------------------------------------
__global__ void __launch_bounds__(256)
reduce_kernel(const float* __restrict__ partial,
              const float* __restrict__ bias,
              float* __restrict__ out) {
    int t = blockIdx.x * blockDim.x + threadIdx.x;
    if (t >= B_ROWS * 3) return;
    float s = 0.0f;
    #pragma unroll
    for (int ks = 0; ks < KS; ++ks)
        s += partial[(size_t)ks * (B_ROWS * 3) + t];
    out[t] = s + (float)N_PTS * bias[t % 3];
}

// ---------------------------------------------------------------------
extern "C" void kernel_launch(void* const* d_in, const int* in_sizes, int n_in,
                              void* d_out, int out_size, void* d_ws, size_t ws_size,
                              hipStream_t stream) {
    const float* x    = (const float*)d_in[0];   // [4096, 24576] fp32
    const float* w    = (const float*)d_in[1];   // [8192, 6]     fp32
    const float* bias = (const float*)d_in[2];   // [3]           fp32
    float*       out  = (float*)d_out;           // [4096, 3]     fp32

    // workspace layout
    float* bpack   = (float*)d_ws;                       // K_TOT*16 floats (1.57 MB)
    float* partial = bpack + (size_t)K_TOT * 16;         // KS*B_ROWS*3 floats (1.57 MB)

    prep_kernel<<<N_PTS / 256, 256, 0, stream>>>(w, bpack);

    const int total_waves = M_TILES * KS;                // 8192 waves
    gemm_kernel<<<total_waves / WAVES_PER_BLOCK, 256, 0, stream>>>(x, bpack, partial);

    reduce_kernel<<<(B_ROWS * 3 + 255) / 256, 256, 0, stream>>>(partial, bias, out);
}